// LSTMCell_81870666597040
// MI455X (gfx1250) — compile-verified
//
#include <hip/hip_runtime.h>
#include <cmath>

typedef __attribute__((ext_vector_type(2))) float v2f;
typedef __attribute__((ext_vector_type(8))) float v8f;

#define B_   64
#define CIN_ 256
#define H_   256
#define T_   512
#define G4_  1024   // 4*H

// ---------------------------------------------------------------------------
// Phase 1: WX[b,t,j] = sum_i x[b,i,t] * W_ih[b,i,j]   (fp32 WMMA 16x16x4)
// grid.x = B*(T/16), grid.y = 16 ; block = 128 (4 waves); wave w -> j-tile
// A tile (16x4, f32): lane m=lane&15 holds M=m; VGPR0/1 = K {0,1} (lanes<16)
//                     or K {2,3} (lanes>=16).  B tile mirrors this with N.
// ---------------------------------------------------------------------------
__global__ __launch_bounds__(128) void wx_gemm(const float* __restrict__ x,
                                               const float* __restrict__ Wih,
                                               float* __restrict__ WX) {
  const int lane = threadIdx.x & 31;
  const int wave = threadIdx.x >> 5;
  const int b    = blockIdx.x >> 5;            // T/16 = 32 t-tiles per batch
  const int t0   = (blockIdx.x & 31) << 4;
  const int j0   = ((blockIdx.y << 2) + wave) << 4;

  const int mn = lane & 15;                    // M (for A) and N (for B)
  const int kb = (lane >> 4) << 1;             // 0 or 2

  // x is (B, CIN, T): element [b, i, t0+mn] ; W_ih is (B, CIN, 4H): [b, i, j0+mn]
  const float* xa = x   + (size_t)b * CIN_ * T_  + t0 + mn;
  const float* wb = Wih + (size_t)b * CIN_ * G4_ + j0 + mn;

  v8f acc0 = {};
  v8f acc1 = {};
#pragma unroll 4
  for (int i0 = 0; i0 < CIN_; i0 += 8) {
    v2f a0, b0, a1, b1;
    a0.x = xa[(size_t)(i0 + kb    ) * T_ ];
    a0.y = xa[(size_t)(i0 + kb + 1) * T_ ];
    b0.x = wb[(size_t)(i0 + kb    ) * G4_];
    b0.y = wb[(size_t)(i0 + kb + 1) * G4_];
    a1.x = xa[(size_t)(i0 + 4 + kb    ) * T_ ];
    a1.y = xa[(size_t)(i0 + 4 + kb + 1) * T_ ];
    b1.x = wb[(size_t)(i0 + 4 + kb    ) * G4_];
    b1.y = wb[(size_t)(i0 + 4 + kb + 1) * G4_];
    acc0 = __builtin_amdgcn_wmma_f32_16x16x4_f32(false, a0, false, b0,
                                                 (short)0, acc0, false, false);
    acc1 = __builtin_amdgcn_wmma_f32_16x16x4_f32(false, a1, false, b1,
                                                 (short)0, acc1, false, false);
  }
  v8f acc = acc0 + acc1;

  // D layout: VGPR v holds M = v (lanes 0-15) or v+8 (lanes 16-31), N = lane&15
  float* out = WX + (size_t)b * T_ * G4_ + j0 + mn;
  const int mh = (lane >> 4) << 3;             // 0 or 8
#pragma unroll
  for (int v = 0; v < 8; ++v) {
    out[(size_t)(t0 + mh + v) * G4_] = acc[v];
  }
}

// ---------------------------------------------------------------------------
// Phase 2: sequential scan. One 1024-thread block (32 waves = full WGP) per
// batch. Thread j owns gate column j of 4H; W_hh rows stream fully coalesced
// from L2 (whole W_hh set = 67 MB, resident in the 192 MB L2 across steps).
// ---------------------------------------------------------------------------
__device__ __forceinline__ float sigm(float v) { return 1.0f / (1.0f + __expf(-v)); }

__global__ __launch_bounds__(1024) void lstm_scan(const float* __restrict__ h0,
                                                  const float* __restrict__ c0,
                                                  const float* __restrict__ Whh,
                                                  const float* __restrict__ bh,
                                                  const float* __restrict__ WX,
                                                  float* __restrict__ h_out,
                                                  float* __restrict__ c_out) {
  __shared__ float hvec[H_];
  __shared__ float gates[G4_];

  const int b = blockIdx.x;
  const int j = threadIdx.x;                   // 0..1023

  const float bias        = bh[(size_t)b * G4_ + j];
  const float* __restrict__ wcol = Whh + (size_t)b * H_ * G4_ + j;
  const float* __restrict__ wxp  = WX  + (size_t)b * T_ * G4_ + j;

  float creg = 0.0f;
  if (j < H_) {
    hvec[j] = h0[(size_t)b * H_ + j];
    creg    = c0[(size_t)b * H_ + j];
  }
  __syncthreads();

  for (int t = 0; t < T_; ++t) {
    // wh_j = sum_h hvec[h] * W_hh[b][h][j]  (rows coalesced across the block)
    float acc = 0.0f;
#pragma unroll 4
    for (int h = 0; h < H_; h += 4) {
      float4 hv = *reinterpret_cast<const float4*>(&hvec[h]);
      acc = __builtin_fmaf(hv.x, wcol[(size_t)(h + 0) * G4_], acc);
      acc = __builtin_fmaf(hv.y, wcol[(size_t)(h + 1) * G4_], acc);
      acc = __builtin_fmaf(hv.z, wcol[(size_t)(h + 2) * G4_], acc);
      acc = __builtin_fmaf(hv.w, wcol[(size_t)(h + 3) * G4_], acc);
    }
    gates[j] = tanhf(wxp[(size_t)t * G4_] + acc + bias);
    __syncthreads();

    if (j < H_) {
      float ig = gates[j];
      float fg = gates[H_ + j];
      float gg = gates[2 * H_ + j];
      float og = gates[3 * H_ + j];
      creg = creg * sigm(fg) + sigm(ig) * tanhf(gg);
      float hn = sigm(og) * tanhf(creg);
      hvec[j] = hn;
      h_out[((size_t)b * H_ + j) * T_ + t] = hn;   // h_out is (B, H, T)
    }
    __syncthreads();
  }
  if (j < H_) c_out[(size_t)b * H_ + j] = creg;
}

// ---------------------------------------------------------------------------
extern "C" void kernel_launch(void* const* d_in, const int* in_sizes, int n_in,
                              void* d_out, int out_size, void* d_ws, size_t ws_size,
                              hipStream_t stream) {
  const float* x   = (const float*)d_in[0];
  const float* h0  = (const float*)d_in[1];
  const float* c0  = (const float*)d_in[2];
  const float* Wih = (const float*)d_in[3];
  const float* Whh = (const float*)d_in[4];
  const float* bh  = (const float*)d_in[5];

  float* WX    = (float*)d_ws;                       // B*T*4H*4 = 134,217,728 B
  float* h_out = (float*)d_out;                      // (B, H, T)
  float* c_out = h_out + (size_t)B_ * H_ * T_;       // (B, 1, H) appended

  dim3 g1(B_ * (T_ / 16), 16);
  wx_gemm<<<g1, 128, 0, stream>>>(x, Wih, WX);
  lstm_scan<<<B_, 1024, 0, stream>>>(h0, c0, Whh, bh, WX, h_out, c_out);
}